// EMD_58351425683471
// MI455X (gfx1250) — compile-verified
//
#include <hip/hip_runtime.h>

// MI455X / gfx1250 approximate-EMD (auction) kernel.
//
// One workgroup per batch (8 total), 1024 threads = 32 wave32 waves.
// All auction state lives in LDS (~124 KB << 320 KB/WGP). The pairwise cost
// is never materialized: each bidding pass recomputes value tiles with
// V_WMMA_F32_16X16X4_F32, with the column price folded into the K=4 B-operand:
//   A row i  = (2x0, 2x1, 2x2, 1)
//   B col j  = (y0, y1, y2, -(|y_j|^2 + price_j))
//   D[i][j]  = 2 x.y - |y|^2 - price = value_ij + |x_i|^2   (row-constant
//              offset cancels in the top-2 gap, so it never matters)
//
// Round-2 changes: fully branch-free top-2 update and merge (v_max/v_min +
// v_cndmask instead of EXEC-mask control flow), software-pipelined B-operand
// ds_load_b64 (double buffer), tile loop unrolled x4.

typedef __attribute__((ext_vector_type(2))) float v2f;
typedef __attribute__((ext_vector_type(8))) float v8f;

#define NPTS   2048
#define NBATCH 8
#define AITERS 20
#define AEPS   0.02f
#define BLOCK  1024
#define WAVES  (BLOCK / 32)            // 32 waves
#define ROWS_PER_WAVE (NPTS / WAVES)   // 64
#define STRIPS (ROWS_PER_WAVE / 16)    // 4 strips of 16 rows per wave
#define NTILES (NPTS / 16)             // 128 column tiles

__global__ __launch_bounds__(BLOCK) void emd_auction_kernel(
    const float* __restrict__ x, const float* __restrict__ y,
    float* __restrict__ per_batch)
{
    __shared__ float xp[NPTS * 4];                // {2x0,2x1,2x2,1}
    __shared__ float yp[NPTS * 4];                // {y0,y1,y2,-(y2+price)}
    __shared__ float y2s[NPTS];
    __shared__ float price[NPTS];
    __shared__ int   owner[NPTS];
    __shared__ unsigned long long colKey[NPTS];   // packed (bid_bits, ~i)
    __shared__ int   assignedf[NPTS];
    __shared__ int   arow[NPTS];
    __shared__ float red[BLOCK];
    __shared__ int   unassignedCnt;

    const int b       = blockIdx.x;
    const int tid     = threadIdx.x;
    const int lane    = tid & 31;
    const int wave    = tid >> 5;
    const int lane_lo = lane & 15;
    const int half    = lane >> 4;

    const float* xb = x + (size_t)b * NPTS * 3;
    const float* yb = y + (size_t)b * NPTS * 3;

    // Warm L2 (emits global_prefetch_b8 on gfx1250).
    __builtin_prefetch(xb + tid, 0, 1);
    __builtin_prefetch(yb + tid, 0, 1);

    // ---- init LDS state ----
    for (int i = tid; i < NPTS; i += BLOCK) {
        float x0 = xb[i * 3 + 0], x1 = xb[i * 3 + 1], x2 = xb[i * 3 + 2];
        xp[i * 4 + 0] = 2.0f * x0;
        xp[i * 4 + 1] = 2.0f * x1;
        xp[i * 4 + 2] = 2.0f * x2;
        xp[i * 4 + 3] = 1.0f;
        float y0 = yb[i * 3 + 0], y1 = yb[i * 3 + 1], y2 = yb[i * 3 + 2];
        float ysq = y0 * y0 + y1 * y1 + y2 * y2;
        yp[i * 4 + 0] = y0;
        yp[i * 4 + 1] = y1;
        yp[i * 4 + 2] = y2;
        yp[i * 4 + 3] = -ysq;            // price == 0 initially
        y2s[i]   = ysq;
        price[i] = 0.0f;
        owner[i] = -1;
        colKey[i] = 0ULL;
    }
    __syncthreads();

    // ---- Jacobi auction iterations ----
    for (int it = 0; it < AITERS; ++it) {
        // phase 1: assigned flags from owner map
        for (int i = tid; i < NPTS; i += BLOCK) assignedf[i] = 0;
        if (tid == 0) unassignedCnt = 0;
        __syncthreads();
        for (int j = tid; j < NPTS; j += BLOCK) {
            int o = owner[j];
            if (o >= 0) assignedf[o] = 1;
        }
        __syncthreads();
        int localUn = 0;
        for (int i = tid; i < NPTS; i += BLOCK) localUn += (assignedf[i] == 0);
        if (localUn) atomicAdd(&unassignedCnt, localUn);
        __syncthreads();
        if (unassignedCnt == 0) break;   // converged: nothing changes anymore

        // phase 2: bidding — WMMA value tiles + branch-free per-row top-2
        for (int s = 0; s < STRIPS; ++s) {
            const int rowBase = wave * ROWS_PER_WAVE + s * 16;
            int myAssigned = assignedf[rowBase + lane_lo];
            if (__all(myAssigned)) continue;   // wave-uniform strip skip

            // A operand (16x4 f32): lanes 0-15 hold {K0,K1}, 16-31 {K2,K3}
            v2f Aop = *(const v2f*)&xp[(rowBase + lane_lo) * 4 + half * 2];

            float v1[8], v2r[8];
            int   j1[8], j2[8];
#pragma unroll
            for (int r = 0; r < 8; ++r) {
                v1[r] = -3.0e38f; v2r[r] = -3.0e38f; j1[r] = 0; j2[r] = 0;
            }

            // software-pipelined B-operand feed (double buffer)
            const int bOff = lane_lo * 4 + half * 2;
            v2f Bcur = *(const v2f*)&yp[bOff];

#pragma unroll 4
            for (int t = 0; t < NTILES; ++t) {
                v2f Bnext = (t + 1 < NTILES)
                                ? *(const v2f*)&yp[(t + 1) * 64 + bOff]
                                : Bcur;
                v8f c = {0.f, 0.f, 0.f, 0.f, 0.f, 0.f, 0.f, 0.f};
                v8f d = __builtin_amdgcn_wmma_f32_16x16x4_f32(
                    false, Aop, false, Bcur, (short)0, c, false, false);
                const int jcol = t * 16 + lane_lo;
#pragma unroll
                for (int r = 0; r < 8; ++r) {
                    float v   = d[r];
                    bool  g1  = v > v1[r];
                    bool  g2  = v > v2r[r];
                    // new top2 value = max(old top2, min(v, old top1))
                    float nv2 = fmaxf(v2r[r], fminf(v, v1[r]));
                    int   nj2 = g2 ? (g1 ? j1[r] : jcol) : j2[r];
                    float nv1 = fmaxf(v, v1[r]);
                    int   nj1 = g1 ? jcol : j1[r];
                    v1[r] = nv1; j1[r] = nj1;
                    v2r[r] = nv2; j2[r] = nj2;
                }
                Bcur = Bnext;
            }

            // branch-free cross-lane top-2 merge inside each 16-lane half
#pragma unroll
            for (int r = 0; r < 8; ++r) {
#pragma unroll
                for (int m = 8; m >= 1; m >>= 1) {
                    float w1 = __shfl_xor(v1[r], m);
                    int   k1 = __shfl_xor(j1[r], m);
                    float w2 = __shfl_xor(v2r[r], m);
                    int   k2 = __shfl_xor(j2[r], m);
                    bool  t1 = w1 > v1[r];               // remote wins top1
                    float nv1 = t1 ? w1 : v1[r];
                    int   nj1 = t1 ? k1 : j1[r];
                    float lo  = t1 ? v1[r] : w1;         // loser of top1s
                    int   loj = t1 ? j1[r] : k1;
                    float s2  = t1 ? w2 : v2r[r];        // winner side's top2
                    int   s2j = t1 ? k2 : j2[r];
                    bool  u   = lo > s2;
                    v1[r]  = nv1;             j1[r] = nj1;
                    v2r[r] = u ? lo : s2;     j2[r] = u ? loj : s2j;
                }
            }

            // one lane per half places each row's bid
#pragma unroll
            for (int r = 0; r < 8; ++r) {
                if (lane_lo == r) {
                    int row = rowBase + r + half * 8;
                    if (!assignedf[row]) {
                        float incr = v1[r] - v2r[r] + AEPS;   // x2 offset cancels
                        float bid  = price[j1[r]] + incr;     // always > 0
                        unsigned int bb = __float_as_uint(bid);
                        unsigned long long key =
                            ((unsigned long long)bb << 32) |
                            (unsigned int)(0x7FFFFFFF - row); // low i wins ties
                        atomicMax(&colKey[j1[r]], key);       // ds_max_u64
                    }
                }
            }
        }
        __syncthreads();

        // phase 3: resolve winners, update price/owner, refresh B K=3 slot
        for (int j = tid; j < NPTS; j += BLOCK) {
            unsigned long long k = colKey[j];
            if (k != 0ULL) {
                float nb = __uint_as_float((unsigned int)(k >> 32));
                price[j] = nb;
                owner[j] = 0x7FFFFFFF - (int)(unsigned int)(k & 0xFFFFFFFFu);
                yp[j * 4 + 3] = -(y2s[j] + nb);
                colKey[j] = 0ULL;
            }
        }
        __syncthreads();
    }

    // ---- invert owner map, NN fallback, gather distances ----
    for (int i = tid; i < NPTS; i += BLOCK) arow[i] = -1;
    __syncthreads();
    for (int j = tid; j < NPTS; j += BLOCK) {
        int o = owner[j];
        if (o >= 0) arow[o] = j;
    }
    __syncthreads();

    float lsum = 0.0f;
    for (int i = tid; i < NPTS; i += BLOCK) {
        float a0 = xp[i * 4 + 0], a1 = xp[i * 4 + 1], a2 = xp[i * 4 + 2]; // 2x
        int aj = arow[i];
        if (aj < 0) {
            // nearest neighbor: argmin_j (|y|^2 - 2 x.y); branch-free select
            float best = 3.0e38f;
            int   bj   = 0;
            for (int j = 0; j < NPTS; ++j) {
                float d = y2s[j] - (a0 * yp[j * 4 + 0] +
                                    a1 * yp[j * 4 + 1] +
                                    a2 * yp[j * 4 + 2]);
                bool lt = d < best;
                best = lt ? d : best;
                bj   = lt ? j : bj;
            }
            aj = bj;
        }
        float xsq = 0.25f * (a0 * a0 + a1 * a1 + a2 * a2);
        float cst = xsq + y2s[aj] - (a0 * yp[aj * 4 + 0] +
                                     a1 * yp[aj * 4 + 1] +
                                     a2 * yp[aj * 4 + 2]);
        lsum += cst;
    }

    // deterministic block tree-reduction
    red[tid] = lsum;
    __syncthreads();
    for (int s = BLOCK / 2; s > 0; s >>= 1) {
        if (tid < s) red[tid] += red[tid + s];
        __syncthreads();
    }
    if (tid == 0) per_batch[b] = red[0] * (1.0f / (float)NPTS);
}

__global__ void emd_finalize_kernel(const float* __restrict__ per_batch,
                                    float* __restrict__ out)
{
    if (blockIdx.x == 0 && threadIdx.x == 0) {
        float s = 0.0f;
        for (int b = 0; b < NBATCH; ++b) s += per_batch[b];  // fixed order
        out[0] = s * (1.0f / (float)NBATCH);
    }
}

extern "C" void kernel_launch(void* const* d_in, const int* in_sizes, int n_in,
                              void* d_out, int out_size, void* d_ws, size_t ws_size,
                              hipStream_t stream)
{
    (void)in_sizes; (void)n_in; (void)out_size; (void)ws_size;
    const float* x = (const float*)d_in[0];
    const float* y = (const float*)d_in[1];
    float* per_batch = (float*)d_ws;      // 8 floats of scratch
    float* out = (float*)d_out;

    emd_auction_kernel<<<NBATCH, BLOCK, 0, stream>>>(x, y, per_batch);
    emd_finalize_kernel<<<1, 32, 0, stream>>>(per_batch, out);
}